// SoftmaxAggregator_38663295598909
// MI455X (gfx1250) — compile-verified
//
#include <hip/hip_runtime.h>
#include <math.h>

#define DEV __device__ __forceinline__

// ---------------------------------------------------------------------------
// Segment boundaries: starts[t] = lower_bound(idx, N, t), for t in [0, S].
// idx is sorted (reference does jnp.sort), so segment s occupies rows
// [starts[s], starts[s+1]).
// ---------------------------------------------------------------------------
DEV int lower_bound_dev(const int* __restrict__ a, int n, int key) {
  int lo = 0, hi = n;
  while (lo < hi) {
    int mid = (lo + hi) >> 1;
    if (a[mid] < key) lo = mid + 1; else hi = mid;
  }
  return lo;
}

__global__ __launch_bounds__(256) void seg_bounds_kernel(
    const int* __restrict__ idx, int* __restrict__ starts, int N, int S) {
  int t = blockIdx.x * blockDim.x + threadIdx.x;
  if (t > S) return;
  starts[t] = lower_bound_dev(idx, N, t);
}

// ---------------------------------------------------------------------------
// Online-softmax update for one feature.
// nm = max(m, z); exactly one of {scale, e} is exp(0)=1, so a single
// v_exp_f32 with free -|x| source modifiers suffices: t = exp(-|m - z|).
// First iteration (m = -inf): dm = -inf -> t = 0, mkeep = false
//   => e = 1, scale = 0, nm = z.  Correct.
// ---------------------------------------------------------------------------
DEV void upd1(float& m, float& sm, float& ac, float xv, float beta) {
  float z  = beta * xv;
  float dm = m - z;
  float t  = __expf(-fabsf(dm));      // exp(-|m-z|)
  bool  mkeep = (dm >= 0.0f);         // running max m survives
  float e     = mkeep ? t : 1.0f;
  float scale = mkeep ? 1.0f : t;
  float nm    = mkeep ? m : z;
  sm = fmaf(sm, scale, e);
  ac = fmaf(ac, scale, e * xv);
  m  = nm;
}

DEV void upd4(float4& m, float4& sm, float4& ac, const float4 xv, float beta) {
  upd1(m.x, sm.x, ac.x, xv.x, beta);
  upd1(m.y, sm.y, ac.y, xv.y, beta);
  upd1(m.z, sm.z, ac.z, xv.z, beta);
  upd1(m.w, sm.w, ac.w, xv.w, beta);
}

// ---------------------------------------------------------------------------
// Async-copy one 1KB row (256 floats) of x into an LDS buffer.
// GVS mode: mem_addr = SGPR64(x) + VGPR32(byte offset). Each lane moves 16B;
// 32 lanes x 16B = 512B per instruction -> 2 instructions per row.
// Tracked by ASYNCcnt (gfx1250 async path).
// ---------------------------------------------------------------------------
DEV void async_row_to_lds(const float* xbase, uint32_t gbyte, uint32_t lds_base,
                          int lane) {
  uint32_t va0 = gbyte + (uint32_t)lane * 16u;
  uint32_t la0 = lds_base + (uint32_t)lane * 16u;
  asm volatile("global_load_async_to_lds_b128 %0, %1, %2"
               :: "v"(la0), "v"(va0), "s"(xbase) : "memory");
  uint32_t va1 = va0 + 512u;
  uint32_t la1 = la0 + 512u;
  asm volatile("global_load_async_to_lds_b128 %0, %1, %2"
               :: "v"(la1), "v"(va1), "s"(xbase) : "memory");
}

// ---------------------------------------------------------------------------
// One wave32 per segment. Lane l owns features [8l, 8l+8) of the 256
// (C=4 x D=64) feature vector. Rows are streamed global -> LDS with
// global_load_async_to_lds_b128 (double-buffered, s_wait_asynccnt sync),
// then consumed via ds_load_b128. Single pass over x:
// ~0.57 GB @ 23.3 TB/s => ~24.5 us floor (HBM-bound; WMMA not applicable).
// ---------------------------------------------------------------------------
template <bool HAVE_STARTS>
__global__ __launch_bounds__(256) void seg_softmax_kernel(
    const float* __restrict__ x, const float* __restrict__ beta_p,
    const int* __restrict__ idx, const int* __restrict__ starts,
    float* __restrict__ out, int N, int S) {
  // 8 waves/block, 2 x 1KB double buffer per wave = 16KB LDS.
  __shared__ __align__(16) float lbuf[8 * 2 * 256];

  const int w    = threadIdx.x >> 5;
  const int lane = threadIdx.x & 31;
  const int s    = blockIdx.x * (blockDim.x >> 5) + w;
  if (s >= S) return;

  int lo, hi;
  if (HAVE_STARTS) {
    lo = starts[s];
    hi = starts[s + 1];
  } else {
    lo = lower_bound_dev(idx, N, s);
    hi = lower_bound_dev(idx, N, s + 1);
  }

  const float beta = beta_p[0];
  const int f0 = lane << 3;      // feature offset within 256 (multiple of 8)
  const int c  = f0 >> 6;        // quaternion component 0..3 (8 | 64)
  const int d  = f0 & 63;        // feature within component

  float* buf0 = &lbuf[(w * 2 + 0) * 256];
  float* buf1 = &lbuf[(w * 2 + 1) * 256];
  // 32-bit LDS byte offsets for the async destination VGPR.
  const uint32_t lds0 = (uint32_t)(uintptr_t)(void*)buf0;
  const uint32_t lds1 = (uint32_t)(uintptr_t)(void*)buf1;

  const float inf = __builtin_inff();
  float4 m0 = {-inf, -inf, -inf, -inf}, m1 = {-inf, -inf, -inf, -inf};
  float4 s0 = {0.f, 0.f, 0.f, 0.f},    s1 = {0.f, 0.f, 0.f, 0.f};
  float4 a0 = {0.f, 0.f, 0.f, 0.f},    a1 = {0.f, 0.f, 0.f, 0.f};

  if (lo < hi)
    async_row_to_lds(x, (uint32_t)lo * 1024u, lds0, lane);

  for (int n = lo; n < hi; ++n) {
    if (n + 2 < hi)
      __builtin_prefetch(x + (size_t)(n + 2) * 256 + f0, 0, 0);

    const int b = (n - lo) & 1;
    if (n + 1 < hi) {
      // Issue next row into the other buffer, then wait until only those
      // 2 fresh async ops remain (in-order completion => row n is in LDS).
      async_row_to_lds(x, (uint32_t)(n + 1) * 1024u, b ? lds0 : lds1, lane);
      asm volatile("s_wait_asynccnt 0x2" ::: "memory");
    } else {
      asm volatile("s_wait_asynccnt 0x0" ::: "memory");
    }

    const float4* lp =
        reinterpret_cast<const float4*>((b ? buf1 : buf0) + f0);
    float4 c0 = lp[0];
    float4 c1 = lp[1];
    upd4(m0, s0, a0, c0, beta);
    upd4(m1, s1, a1, c1, beta);
  }

  // Epilogue: out = acc / sum (sum >= 1 for non-empty segments, since the
  // running-max element contributes e = 1). Empty segments -> 0, matching
  // jax segment_sum identity. Output layout is transposed: [4, S, D].
  float4 r0 = {0.f, 0.f, 0.f, 0.f}, r1 = {0.f, 0.f, 0.f, 0.f};
  if (hi > lo) {
    r0.x = a0.x / s0.x; r0.y = a0.y / s0.y; r0.z = a0.z / s0.z; r0.w = a0.w / s0.w;
    r1.x = a1.x / s1.x; r1.y = a1.y / s1.y; r1.z = a1.z / s1.z; r1.w = a1.w / s1.w;
  }
  float* op = out + ((size_t)c * S + s) * 64 + d;
  *reinterpret_cast<float4*>(op)     = r0;
  *reinterpret_cast<float4*>(op + 4) = r1;
}

// ---------------------------------------------------------------------------
// Inputs (setup_inputs order): x [N,4,64] f32, beta f32 scalar, idx [N] i32,
// dim i32 (unused), dim_size i32 (S derived from out_size instead).
// Output: [4, S, 64] f32.
// ---------------------------------------------------------------------------
extern "C" void kernel_launch(void* const* d_in, const int* in_sizes, int n_in,
                              void* d_out, int out_size, void* d_ws, size_t ws_size,
                              hipStream_t stream) {
  const float* x      = (const float*)d_in[0];
  const float* beta_p = (const float*)d_in[1];
  const int*   idx    = (const int*)d_in[2];
  float*       out    = (float*)d_out;

  const int N = in_sizes[0] / 256;   // rows of x (each row = 4*64 floats)
  const int S = out_size   / 256;    // destination nodes
  if (S <= 0) return;

  const int WPB = 8;                 // wave32s per 256-thread block
  dim3 blk(256);
  dim3 grd((S + WPB - 1) / WPB);

  if (d_ws != nullptr && ws_size >= (size_t)(S + 1) * sizeof(int)) {
    int* starts = (int*)d_ws;
    seg_bounds_kernel<<<(S + 256) / 256, 256, 0, stream>>>(idx, starts, N, S);
    seg_softmax_kernel<true><<<grd, blk, 0, stream>>>(x, beta_p, idx, starts, out, N, S);
  } else {
    seg_softmax_kernel<false><<<grd, blk, 0, stream>>>(x, beta_p, idx, nullptr, out, N, S);
  }
}